// Model_83519934038414
// MI455X (gfx1250) — compile-verified
//
#include <hip/hip_runtime.h>
#include <stdint.h>

typedef __attribute__((ext_vector_type(2))) float v2f;
typedef __attribute__((ext_vector_type(8))) float v8f;

#define THREADS 256
static constexpr int kN  = 100000;    // input nodes
static constexpr int kE  = 1600000;   // directed edges
static constexpr int kC1 = 262144;    // 64^3 cells after pool1
static constexpr int kC2 = 4096;      // 16^3 cells after pool2
static constexpr int kWtStride = 76;  // LDS row stride: even (b64-aligned), conflict-free

// ---------- order-preserving float<->uint encoding for atomic max ----------
__device__ __forceinline__ unsigned encf(float f){
  unsigned u = __float_as_uint(f);
  return (u & 0x80000000u) ? ~u : (u | 0x80000000u);
}
__device__ __forceinline__ float decf(unsigned u){
  unsigned v = (u & 0x80000000u) ? (u ^ 0x80000000u) : ~u;
  return __uint_as_float(v);
}
// cell-id remaps (pure arithmetic, grids 64 -> 16 -> 4)
__device__ __forceinline__ int cid2of(int c){
  return (((c >> 12) >> 2) * 16 + (((c >> 6) & 63) >> 2)) * 16 + ((c & 63) >> 2);
}
__device__ __forceinline__ int cid3of(int c){
  return (((c >> 8) >> 2) * 4 + (((c >> 4) & 15) >> 2)) * 4 + ((c & 15) >> 2);
}
// node position component per hierarchy level
__device__ __forceinline__ float poscomp(int level, const float* __restrict__ nodes,
                                         int r, int axis){
  if (level == 0) return nodes[r * 3 + axis];
  if (level == 1)
    return (float)((axis == 0) ? (r >> 12) : (axis == 1) ? ((r >> 6) & 63) : (r & 63));
  return (float)((axis == 0) ? (r >> 8) : (axis == 1) ? ((r >> 4) & 15) : (r & 15));
}

// ---------- kernels ----------
__global__ void k_cid1(const float* __restrict__ nodes, int* __restrict__ cid, int n){
  int i = blockIdx.x * blockDim.x + threadIdx.x;
  if (i >= n) return;
  int cx = min(max((int)floorf(nodes[i * 3 + 0] * 0.25f), 0), 63);
  int cy = min(max((int)floorf(nodes[i * 3 + 1] * 0.25f), 0), 63);
  int cz = min(max((int)floorf(nodes[i * 3 + 2] * 0.25f), 0), 63);
  cid[i] = (cx * 64 + cy) * 64 + cz;
}

__global__ void k_init4(uint4* __restrict__ t, long long n4){
  long long i = (long long)blockIdx.x * blockDim.x + threadIdx.x;
  if (i < n4) t[i] = make_uint4(0u, 0u, 0u, 0u);
}

// Xcat[r, 0:kPad] = [feats(r) ; pos(r) ; zero-pad]  (one block per row)
__global__ void k_prep(int level, const float* __restrict__ nodes,
                       const float* __restrict__ feats, int in_dim, int kPad,
                       float* __restrict__ Xcat){
  int r = blockIdx.x;
  int k = threadIdx.x;
  if (k >= kPad) return;
  float v = 0.f;
  if (k < in_dim)          v = feats[(long long)r * in_dim + k];
  else if (k < in_dim + 3) v = poscomp(level, nodes, r, k - in_dim);
  Xcat[(long long)r * kPad + k] = v;
}

// a[n, out] = Xcat @ W via exact-f32 WMMA 16x16x4.
// grid.x * 8 waves cover M tiles; grid.y covers 16-col N tiles (shared via LDS).
__global__ void __launch_bounds__(THREADS)
k_gemm(const float* __restrict__ Xcat, int kPad, int kTot,
       const float* __restrict__ W, int out_dim, int n_rows,
       float* __restrict__ aOut){
  __shared__ float Wt[16 * kWtStride];   // transposed 16-col weight slice, zero-padded
  int colBase = blockIdx.y * 16;
  for (int i = threadIdx.x; i < 16 * kWtStride; i += THREADS){
    int cl = i / kWtStride, kk = i - cl * kWtStride;
    int col = colBase + cl;
    float v = (kk < kTot && col < out_dim) ? W[kk * out_dim + col] : 0.f;
    Wt[cl * kWtStride + kk] = v;
  }
  __syncthreads();
  int wave   = threadIdx.x >> 5;
  int lane   = threadIdx.x & 31;
  int tileM  = blockIdx.x * 8 + wave;
  int tilesM = (n_rows + 15) >> 4;
  if (tileM >= tilesM) return;           // wave-uniform; EXEC all-1s for WMMA
  int m    = lane & 15;
  int half = lane >> 4;                  // lanes 0-15: K=k0+{0,1}; 16-31: K=k0+{2,3}
  int row  = tileM * 16 + m;
  int rowc = min(row, n_rows - 1);       // clamp: garbage rows masked at store
  const float* arow = Xcat + (long long)rowc * kPad + half * 2;
  const float* brow = Wt + m * kWtStride + half * 2;
  v8f acc = {};
  for (int k0 = 0; k0 < kPad; k0 += 4){
    v2f A = *(const v2f*)(arow + k0);    // global b64, contiguous in K
    v2f B = *(const v2f*)(brow + k0);    // ds b64
    acc = __builtin_amdgcn_wmma_f32_16x16x4_f32(false, A, false, B, (short)0, acc,
                                                false, false);
  }
  int col = colBase + m;
  if (col < out_dim){
#pragma unroll
    for (int v = 0; v < 8; ++v){
      int orow = tileM * 16 + v + half * 8;
      if (orow < n_rows)
        aOut[(long long)orow * out_dim + col] = acc[v];
    }
  }
}

// per (edge, 16-feature chunk): tmp[dst] max= enc(a[src]); indices loaded once/chunk
__global__ void k_scatter_edges(int level, const int* __restrict__ esrc,
                                const int* __restrict__ edst, long long nthr,
                                int sh, int csh, const int* __restrict__ cid1,
                                const float* __restrict__ a, unsigned* __restrict__ tmp){
  long long idx = (long long)blockIdx.x * blockDim.x + threadIdx.x;
  if (idx >= nthr) return;
  int e     = (int)(idx >> csh);
  int jbase = ((int)(idx & ((1 << csh) - 1))) << 4;
  int s = esrc[e], d = edst[e];
  if (level >= 1){ s = cid1[s]; d = cid1[d]; }
  if (level == 2){ s = cid2of(s); d = cid2of(d); }
  int chunk = min(1 << sh, 16);
  const float4* ap = (const float4*)(a + ((long long)s << sh) + jbase);
  unsigned*     tp = tmp + ((long long)d << sh) + jbase;
  for (int q = 0; q < (chunk >> 2); ++q){
    float4 v = ap[q];
    atomicMax(tp + 4 * q + 0, encf(v.x));
    atomicMax(tp + 4 * q + 1, encf(v.y));
    atomicMax(tp + 4 * q + 2, encf(v.z));
    atomicMax(tp + 4 * q + 3, encf(v.w));
  }
}

// self loops (identity), vectorized x4
__global__ void k_scatter_self(const float* __restrict__ a, unsigned* __restrict__ tmp,
                               long long n4){
  long long i = (long long)blockIdx.x * blockDim.x + threadIdx.x;
  if (i >= n4) return;
  float4 v = ((const float4*)a)[i];
  unsigned* tp = tmp + i * 4;
  atomicMax(tp + 0, encf(v.x));
  atomicMax(tp + 1, encf(v.y));
  atomicMax(tp + 2, encf(v.z));
  atomicMax(tp + 3, encf(v.w));
}

// x[d,j] = relu( dec(tmp) - pos(d)@W_p ); key==0 (no in-edges) -> 0.  x4 vectorized
__global__ void k_finalize(int level, const float* __restrict__ nodes,
                           const float* __restrict__ W, int in_dim, int out_dim, int sh,
                           long long n4, const unsigned* __restrict__ tmp,
                           float* __restrict__ x){
  long long idx = (long long)blockIdx.x * blockDim.x + threadIdx.x;
  if (idx >= n4) return;
  int d  = (int)(idx >> (sh - 2));
  int jq = ((int)(idx & ((1 << (sh - 2)) - 1))) << 2;
  uint4 key = ((const uint4*)tmp)[idx];
  float p0 = poscomp(level, nodes, d, 0);
  float p1 = poscomp(level, nodes, d, 1);
  float p2 = poscomp(level, nodes, d, 2);
  float4 w0 = *(const float4*)(W + (in_dim + 0) * out_dim + jq);
  float4 w1 = *(const float4*)(W + (in_dim + 1) * out_dim + jq);
  float4 w2 = *(const float4*)(W + (in_dim + 2) * out_dim + jq);
  float4 r;
  r.x = key.x ? fmaxf(decf(key.x) - fmaf(p0, w0.x, fmaf(p1, w1.x, p2 * w2.x)), 0.f) : 0.f;
  r.y = key.y ? fmaxf(decf(key.y) - fmaf(p0, w0.y, fmaf(p1, w1.y, p2 * w2.y)), 0.f) : 0.f;
  r.z = key.z ? fmaxf(decf(key.z) - fmaf(p0, w0.z, fmaf(p1, w1.z, p2 * w2.z)), 0.f) : 0.f;
  r.w = key.w ? fmaxf(decf(key.w) - fmaf(p0, w0.w, fmaf(p1, w1.w, p2 * w2.w)), 0.f) : 0.f;
  ((float4*)x)[idx] = r;
}

// voxel-grid max pooling scatter (x4): mode 0 uses cid1[], modes 1/2 arithmetic remap
__global__ void k_pool_scatter(int mode, const int* __restrict__ cid1,
                               const float* __restrict__ x, unsigned* __restrict__ tmp,
                               long long n4, int sh){
  long long idx = (long long)blockIdx.x * blockDim.x + threadIdx.x;
  if (idx >= n4) return;
  int i  = (int)(idx >> (sh - 2));
  int jq = ((int)(idx & ((1 << (sh - 2)) - 1))) << 2;
  int d = (mode == 0) ? cid1[i] : (mode == 1) ? cid2of(i) : cid3of(i);
  float4 v = *(const float4*)(x + ((long long)i << sh) + jq);
  unsigned* tp = tmp + ((long long)d << sh) + jq;
  atomicMax(tp + 0, encf(v.x));
  atomicMax(tp + 1, encf(v.y));
  atomicMax(tp + 2, encf(v.z));
  atomicMax(tp + 3, encf(v.w));
}

__global__ void k_pool_finalize(const unsigned* __restrict__ tmp, float* __restrict__ x,
                                long long n4){
  long long i = (long long)blockIdx.x * blockDim.x + threadIdx.x;
  if (i >= n4) return;
  uint4 k = ((const uint4*)tmp)[i];
  float4 r;
  r.x = k.x ? decf(k.x) : 0.f;
  r.y = k.y ? decf(k.y) : 0.f;
  r.z = k.z ? decf(k.z) : 0.f;
  r.w = k.w ? decf(k.w) : 0.f;
  ((float4*)x)[i] = r;
}

// out[1,100] = xg[1,4096] @ Wl[4096,100] + bl
__global__ void k_linear(const float* __restrict__ xg, const float* __restrict__ Wl,
                         const float* __restrict__ bl, float* __restrict__ out){
  int j = threadIdx.x;
  if (j >= 100) return;
  float acc = bl[j];
  for (int i = 0; i < 4096; ++i) acc = fmaf(xg[i], Wl[i * 100 + j], acc);
  out[j] = acc;
}

// ---------- host orchestration ----------
static inline unsigned nblk(long long n){ return (unsigned)((n + THREADS - 1) / THREADS); }

extern "C" void kernel_launch(void* const* d_in, const int* in_sizes, int n_in,
                              void* d_out, int out_size, void* d_ws, size_t ws_size,
                              hipStream_t stream) {
  (void)in_sizes; (void)n_in; (void)out_size; (void)ws_size;
  const float* d_nodes = (const float*)d_in[0];
  const float* d_feat  = (const float*)d_in[1];
  const int*   d_edges = (const int*)d_in[2];
  const float* W1 = (const float*)d_in[3];
  const float* W2 = (const float*)d_in[4];
  const float* W3 = (const float*)d_in[5];
  const float* W4 = (const float*)d_in[6];
  const float* W5 = (const float*)d_in[7];
  const float* W6 = (const float*)d_in[8];
  const float* W7 = (const float*)d_in[9];
  const float* Wl = (const float*)d_in[10];
  const float* bl = (const float*)d_in[11];
  const int* esrc = d_edges;
  const int* edst = d_edges + kE;

  // bump allocator on d_ws
  char* ws = (char*)d_ws;
  size_t off = 0;
  auto take = [&](size_t bytes) -> void* {
    void* p = ws + off;
    off = (off + bytes + 255) & ~(size_t)255;
    return p;
  };
  const size_t big = (size_t)kC1 * 32 * sizeof(float);   // largest feature tensor
  const size_t xcb = (size_t)kC1 * 40 * sizeof(float);   // largest padded concat
  int*      cid1 = (int*)take((size_t)kN * sizeof(int));
  float*    bufA = (float*)take(big);       // node-level GEMM output a[]
  unsigned* bufT = (unsigned*)take(big);    // encoded atomic-max accumulator
  float*    bufXc= (float*)take(xcb);       // padded concat input for GEMM
  float*    X0   = (float*)take(big);       // feature ping
  float*    X1   = (float*)take(big);       // feature pong

  auto gconv = [&](int level, const float* xin, int in_dim, const float* W,
                   int out_dim, int n_nodes, bool self, float* xout){
    int sh = 0; while ((1 << sh) < out_dim) sh++;
    int kTot = in_dim + 3;
    int kPad = (kTot + 3) & ~3;
    long long nT = (long long)n_nodes << sh;
    k_init4<<<nblk(nT >> 2), THREADS, 0, stream>>>((uint4*)bufT, nT >> 2);
    int pthr = (kPad + 31) & ~31;
    k_prep<<<n_nodes, pthr, 0, stream>>>(level, d_nodes, xin, in_dim, kPad, bufXc);
    int tilesM = (n_nodes + 15) >> 4, tilesN = (out_dim + 15) >> 4;
    dim3 gg((tilesM + 7) / 8, tilesN);
    k_gemm<<<gg, THREADS, 0, stream>>>(bufXc, kPad, kTot, W, out_dim, n_nodes, bufA);
    int csh = sh > 4 ? sh - 4 : 0;
    long long tE = (long long)kE << csh;
    k_scatter_edges<<<nblk(tE), THREADS, 0, stream>>>(level, esrc, edst, tE, sh, csh,
                                                      cid1, bufA, bufT);
    if (self)
      k_scatter_self<<<nblk(nT >> 2), THREADS, 0, stream>>>(bufA, bufT, nT >> 2);
    k_finalize<<<nblk(nT >> 2), THREADS, 0, stream>>>(level, d_nodes, W, in_dim,
                                                      out_dim, sh, nT >> 2, bufT, xout);
  };
  auto pool = [&](int mode, const float* xin, int out_dim, int n_src, int n_dst,
                  float* xout){
    int sh = 0; while ((1 << sh) < out_dim) sh++;
    long long nT = (long long)n_dst << sh;
    long long tS = (long long)n_src << sh;
    k_init4<<<nblk(nT >> 2), THREADS, 0, stream>>>((uint4*)bufT, nT >> 2);
    k_pool_scatter<<<nblk(tS >> 2), THREADS, 0, stream>>>(mode, cid1, xin, bufT,
                                                          tS >> 2, sh);
    k_pool_finalize<<<nblk(nT >> 2), THREADS, 0, stream>>>(bufT, xout, nT >> 2);
  };

  k_cid1<<<nblk(kN), THREADS, 0, stream>>>(d_nodes, cid1, kN);

  gconv(0, d_feat, 1,  W1,  8, kN,  false, X0);   // x1 [N,8]
  gconv(0, X0,     8,  W2, 16, kN,  false, X1);   // x2 [N,16]
  pool (0, X1, 16, kN,  kC1, X0);                 // xP1 [64^3,16]
  gconv(1, X0,    16,  W3, 32, kC1, true,  X1);   // x3 [64^3,32]
  gconv(1, X1,    32,  W4, 32, kC1, true,  X0);   // x4 [64^3,32]
  pool (1, X0, 32, kC1, kC2, X1);                 // xP2 [16^3,32]
  gconv(2, X1,    32,  W5, 64, kC2, true,  X0);   // x5 [16^3,64]
  gconv(2, X0,    64,  W6, 64, kC2, true,  X1);   // x6
  gconv(2, X1,    64,  W7, 64, kC2, true,  X0);   // x7
  pool (2, X0, 64, kC2, 64, X1);                  // xg [64,64]

  k_linear<<<1, 128, 0, stream>>>(X1, Wl, bl, (float*)d_out);
}